// OptimizedComplexUnitaryGCN_53807350284468
// MI455X (gfx1250) — compile-verified
//
#include <hip/hip_runtime.h>
#include <math.h>

typedef __attribute__((ext_vector_type(2))) float v2f;
typedef __attribute__((ext_vector_type(8))) float v8f;

#define NNODES 8192
#define NEDGES (NNODES * 32)
#define WPR    256          // bitset words per row (8192 bits)
#define NGRAPH 64
#define TERMS  8

// ---------------------------------------------------------------------------
// zero fill (u32)
// ---------------------------------------------------------------------------
__global__ void zero_u32_kernel(unsigned* __restrict__ p, int n) {
    int i = blockIdx.x * blockDim.x + threadIdx.x;
    if (i < n) p[i] = 0u;
}

// ---------------------------------------------------------------------------
// edge list -> directed adjacency bitset A and its transpose AT
// ---------------------------------------------------------------------------
__global__ void build_bits_kernel(const int* __restrict__ ei,
                                  unsigned* __restrict__ bitA,
                                  unsigned* __restrict__ bitAT, int E) {
    int e = blockIdx.x * blockDim.x + threadIdx.x;
    if (e >= E) return;
    int s = ei[e];
    int d = ei[E + e];
    atomicOr(&bitA [(long)s * WPR + (d >> 5)], 1u << (d & 31));
    atomicOr(&bitAT[(long)d * WPR + (s >> 5)], 1u << (s & 31));
}

// ---------------------------------------------------------------------------
// per node: 9 smallest symmetric neighbors -> sub[10] -> ba[11][11] + s
// ba stored as float at ba[n*128 + i*11 + j]
// ---------------------------------------------------------------------------
__global__ void build_ba_kernel(const unsigned* __restrict__ bitA,
                                const unsigned* __restrict__ bitAT,
                                float* __restrict__ ba,
                                int* __restrict__ sArr, int n) {
    int node = blockIdx.x * blockDim.x + threadIdx.x;
    if (node >= n) return;
    const unsigned* ra = bitA  + (long)node * WPR;
    const unsigned* rb = bitAT + (long)node * WPR;
    int sub[10];
    sub[0] = node;
    int cnt = 1;
    for (int w = 0; w < WPR && cnt < 10; ++w) {
        unsigned m = ra[w] | rb[w];
        while (m && cnt < 10) {
            int b = __ffs(m) - 1;
            m &= m - 1;
            sub[cnt++] = w * 32 + b;   // ascending ids == top_k of -ids
        }
    }
    for (; cnt < 10; ++cnt) sub[cnt] = n;   // padding == n (invalid)

    float* bap = ba + (long)node * 128;
    int s = 0;
    for (int i = 0; i < 11; ++i) {
        for (int j = 0; j < 11; ++j) {
            float v = 0.0f;
            if (i < 10 && j < 10) {
                int a = sub[i], b = sub[j];
                if (a < n && b < n) {
                    unsigned wv = bitA[(long)a * WPR + (b >> 5)];
                    v = (float)((wv >> (b & 31)) & 1u);
                }
            }
            bap[i * 11 + j] = v;
            if (i == 0 && j >= 1) s += (int)v;
        }
    }
    sArr[node] = s;
}

// ---------------------------------------------------------------------------
// per-node 11x11 complex matrix exponential (Taylor, 8 terms) of -i*ct*A.
// One wave (32 lanes) per node; 8 nodes per 256-thread block; matrices in LDS.
// Outputs per node: u0 = U[0,0], c = sum_{j=1..s} U[0,j]  (U = G / ||G||_F)
// ---------------------------------------------------------------------------
__global__ void expm_kernel(const float* __restrict__ ba,
                            const int* __restrict__ sArr,
                            const float* __restrict__ trp,
                            const float* __restrict__ tip,
                            float* __restrict__ u0c) {
    __shared__ float Af[8][121];
    __shared__ float Tm[8][242];   // current Taylor term (complex)
    __shared__ float Rm[8][242];   // accumulated result (complex)
    const int lane = threadIdx.x & 31;
    const int w    = threadIdx.x >> 5;
    const int node = blockIdx.x * 8 + w;
    // M = c*A with c = -i*(tr + i*ti) = ti - i*tr
    const float ccr =  tip[0];
    const float cci = -trp[0];

    const float* bap = ba + (long)node * 128;
    int eIdx[4], ex[4], ej[4], ne = 0;
    for (int u = 0; u < 4; ++u) {
        int e = lane + 32 * u;
        if (e < 121) { eIdx[ne] = e; ex[ne] = e / 11; ej[ne] = e % 11; ++ne; }
    }
    for (int t = 0; t < ne; ++t) {
        int e = eIdx[t];
        Af[w][e] = bap[e];
        float d = (ex[t] == ej[t]) ? 1.0f : 0.0f;
        Tm[w][2 * e] = d; Tm[w][2 * e + 1] = 0.0f;
        Rm[w][2 * e] = d; Rm[w][2 * e + 1] = 0.0f;
    }
    __syncthreads();

    for (int step = 1; step <= TERMS; ++step) {
        float nr[4], ni[4];
        const float inv = 1.0f / (float)step;
        for (int t = 0; t < ne; ++t) {
            float sr = 0.0f, si = 0.0f;
            const int x = ex[t], j = ej[t];
            for (int k = 0; k < 11; ++k) {
                float a = Af[w][k * 11 + j];
                sr += Tm[w][2 * (x * 11 + k)    ] * a;
                si += Tm[w][2 * (x * 11 + k) + 1] * a;
            }
            nr[t] = (sr * ccr - si * cci) * inv;
            ni[t] = (sr * cci + si * ccr) * inv;
        }
        __syncthreads();
        for (int t = 0; t < ne; ++t) {
            int e = eIdx[t];
            Tm[w][2 * e]     = nr[t];
            Tm[w][2 * e + 1] = ni[t];
            Rm[w][2 * e]     += nr[t];
            Rm[w][2 * e + 1] += ni[t];
        }
        __syncthreads();
    }

    // Frobenius norm of full 11x11 complex G
    float loc = 0.0f;
    for (int t = 0; t < ne; ++t) {
        int e = eIdx[t];
        float a = Rm[w][2 * e], b = Rm[w][2 * e + 1];
        loc += a * a + b * b;
    }
    for (int off = 16; off; off >>= 1) loc += __shfl_xor(loc, off, 32);
    const float invn = 1.0f / (sqrtf(loc) + 1e-8f);

    const int s = sArr[node];
    float u0r = 0.0f, u0i = 0.0f, cnr = 0.0f, cni = 0.0f;
    if (lane < 11) {   // row 0 entries e = 0..10
        float gr = Rm[w][2 * lane]     * invn;
        float gi = Rm[w][2 * lane + 1] * invn;
        if (lane == 0)      { u0r = gr; u0i = gi; }
        else if (lane <= s) { cnr = gr; cni = gi; }
    }
    for (int off = 16; off; off >>= 1) {
        u0r += __shfl_xor(u0r, off, 32);
        u0i += __shfl_xor(u0i, off, 32);
        cnr += __shfl_xor(cnr, off, 32);
        cni += __shfl_xor(cni, off, 32);
    }
    if (lane == 0) {
        float* o = u0c + (long)node * 4;
        o[0] = u0r; o[1] = u0i; o[2] = cnr; o[3] = cni;
    }
}

// ---------------------------------------------------------------------------
// complex GEMM + bias + complex-relu via V_WMMA_F32_16X16X4_F32.
// C = crelu( (Ar + iAi) @ (Wr + iWi)^T + (br + i bi) )
// A: [M,K] row-major, W: [Nn,K] row-major (so B[k][n] = W[n][k]).
// One wave computes one 16x16 output tile; 8 waves/block.
// HASI=false -> purely real input (layer 1), branch-free specialized loop.
// ---------------------------------------------------------------------------
template <bool HASI>
__global__ void cgemm_wmma_kernel(const float* __restrict__ Ar,
                                  const float* __restrict__ Ai,
                                  const float* __restrict__ Wr,
                                  const float* __restrict__ Wi,
                                  const float* __restrict__ br,
                                  const float* __restrict__ bi,
                                  float* __restrict__ Cr,
                                  float* __restrict__ Ci,
                                  int M, int K, int Nn) {
    const int lane   = threadIdx.x & 31;
    const int wv     = threadIdx.x >> 5;
    const int tilesN = Nn >> 4;
    const int tile   = blockIdx.x * 8 + wv;
    const int tm     = (tile / tilesN) << 4;
    const int tn     = (tile % tilesN) << 4;
    const int half   = lane >> 4;     // 0: K pair {k,k+1}, 1: {k+2,k+3}
    const int l16    = lane & 15;
    const long aBase = (long)(tm + l16) * K;   // A row M = l16
    const long bBase = (long)(tn + l16) * K;   // W row (output col N) = l16

    v8f cr = {};
    v8f ci = {};
    for (int k = 0; k < K; k += 4) {
        const int ka = k + half * 2;
        v2f a_r = *(const v2f*)(Ar + aBase + ka);
        v2f b_r = *(const v2f*)(Wr + bBase + ka);
        v2f b_i = *(const v2f*)(Wi + bBase + ka);
        cr = __builtin_amdgcn_wmma_f32_16x16x4_f32(false, a_r, false, b_r,
                                                   (short)0, cr, false, false);
        ci = __builtin_amdgcn_wmma_f32_16x16x4_f32(false, a_r, false, b_i,
                                                   (short)0, ci, false, false);
        if (HASI) {
            v2f a_i = *(const v2f*)(Ai + aBase + ka);
            v2f a_in; a_in[0] = -a_i[0]; a_in[1] = -a_i[1];   // f32 WMMA has no A-neg bit
            cr = __builtin_amdgcn_wmma_f32_16x16x4_f32(false, a_in, false, b_i,
                                                       (short)0, cr, false, false);
            ci = __builtin_amdgcn_wmma_f32_16x16x4_f32(false, a_i, false, b_r,
                                                       (short)0, ci, false, false);
        }
    }
    // C layout: lanes 0-15: (M = tm+v,   N = tn+l16); lanes 16-31: (M = tm+8+v, N)
    const int   col = tn + l16;
    const float bre = br[col];
    const float bim = bi[col];
    for (int v = 0; v < 8; ++v) {
        const int row = tm + half * 8 + v;
        Cr[(long)row * Nn + col] = fmaxf(cr[v] + bre, 0.0f);
        Ci[(long)row * Nn + col] = fmaxf(ci[v] + bim, 0.0f);
    }
}

// ---------------------------------------------------------------------------
// column mean of an [rows,F] matrix pair; grid = 2*F blocks (real then imag)
// ---------------------------------------------------------------------------
__global__ void colmean_kernel(const float* __restrict__ srcR,
                               const float* __restrict__ srcI,
                               float* __restrict__ dstR,
                               float* __restrict__ dstI,
                               int rows, int F) {
    __shared__ float sh[256];
    const int b = blockIdx.x;
    const float* src = (b < F) ? srcR : srcI;
    float*       dst = (b < F) ? dstR : dstI;
    const int col = (b < F) ? b : b - F;
    float acc = 0.0f;
    for (int r = threadIdx.x; r < rows; r += 256) acc += src[(long)r * F + col];
    sh[threadIdx.x] = acc;
    __syncthreads();
    for (int s = 128; s > 0; s >>= 1) {
        if (threadIdx.x < s) sh[threadIdx.x] += sh[threadIdx.x + s];
        __syncthreads();
    }
    if (threadIdx.x == 0) dst[col] = sh[0] / (float)rows;
}

// ---------------------------------------------------------------------------
// ev = u0*h + c*mean_h + eb  (complex) ; writes H
// ---------------------------------------------------------------------------
__global__ void ev_kernel(const float* __restrict__ Gr, const float* __restrict__ Gi,
                          const float* __restrict__ u0c,
                          const float* __restrict__ mr, const float* __restrict__ mi,
                          const float* __restrict__ ebr, const float* __restrict__ ebi,
                          float* __restrict__ Hr, float* __restrict__ Hi, int F) {
    long idx = (long)blockIdx.x * blockDim.x + threadIdx.x;
    int n = (int)(idx / F);
    int f = (int)(idx % F);
    const float* u = u0c + (long)n * 4;
    float u0r = u[0], u0i = u[1], cre = u[2], cim = u[3];
    float gr = Gr[idx], gi = Gi[idx];
    float mrr = mr[f], mii = mi[f];
    Hr[idx] = u0r * gr - u0i * gi + cre * mrr - cim * mii + ebr[f];
    Hi[idx] = u0r * gi + u0i * gr + cre * mii + cim * mrr + ebi[f];
}

// ---------------------------------------------------------------------------
// same as ev_kernel but writes |ev| (final layer, feeds pooling)
// ---------------------------------------------------------------------------
__global__ void ev_abs_kernel(const float* __restrict__ Gr, const float* __restrict__ Gi,
                              const float* __restrict__ u0c,
                              const float* __restrict__ mr, const float* __restrict__ mi,
                              const float* __restrict__ ebr, const float* __restrict__ ebi,
                              float* __restrict__ xa, int F) {
    long idx = (long)blockIdx.x * blockDim.x + threadIdx.x;
    int n = (int)(idx / F);
    int f = (int)(idx % F);
    const float* u = u0c + (long)n * 4;
    float u0r = u[0], u0i = u[1], cre = u[2], cim = u[3];
    float gr = Gr[idx], gi = Gi[idx];
    float mrr = mr[f], mii = mi[f];
    float evr = u0r * gr - u0i * gi + cre * mrr - cim * mii + ebr[f];
    float evi = u0r * gi + u0i * gr + cre * mii + cim * mrr + ebi[f];
    xa[idx] = sqrtf(evr * evr + evi * evi);
}

// ---------------------------------------------------------------------------
// segment-mean pooling (batch is sorted -> contiguous ranges, no atomics)
// fused with log_softmax. One block (128 threads) per graph.
// ---------------------------------------------------------------------------
__global__ void pool_softmax_kernel(const float* __restrict__ xa,
                                    const int* __restrict__ batch,
                                    float* __restrict__ out, int n, int F) {
    __shared__ float sh[128];
    __shared__ int range[2];
    const int g = blockIdx.x;
    const int t = threadIdx.x;
    if (t == 0) {
        int lo = 0, hi = n;
        while (lo < hi) { int m = (lo + hi) >> 1; if (batch[m] < g) lo = m + 1; else hi = m; }
        range[0] = lo;
        hi = n;
        while (lo < hi) { int m = (lo + hi) >> 1; if (batch[m] < g + 1) lo = m + 1; else hi = m; }
        range[1] = lo;
    }
    __syncthreads();
    const int s0 = range[0], s1 = range[1];
    float acc = 0.0f;
    for (int r = s0; r < s1; ++r) acc += xa[(long)r * F + t];
    const float cntf = fmaxf((float)(s1 - s0), 1.0f);
    const float v = acc / cntf;
    sh[t] = v;
    __syncthreads();
    for (int s = 64; s > 0; s >>= 1) {
        if (t < s) sh[t] = fmaxf(sh[t], sh[t + s]);
        __syncthreads();
    }
    const float mx = sh[0];
    __syncthreads();
    sh[t] = expf(v - mx);
    __syncthreads();
    for (int s = 64; s > 0; s >>= 1) {
        if (t < s) sh[t] += sh[t + s];
        __syncthreads();
    }
    out[(long)g * F + t] = v - (mx + logf(sh[0]));
}

// ---------------------------------------------------------------------------
extern "C" void kernel_launch(void* const* d_in, const int* in_sizes, int n_in,
                              void* d_out, int out_size, void* d_ws, size_t ws_size,
                              hipStream_t stream) {
    (void)in_sizes; (void)n_in; (void)out_size; (void)ws_size;
    const int N = NNODES, E = NEDGES;

    const float* x    = (const float*)d_in[0];
    const float* Wr1  = (const float*)d_in[1];
    const float* Wi1  = (const float*)d_in[2];
    const float* br1  = (const float*)d_in[3];
    const float* bi1  = (const float*)d_in[4];
    const float* tr1  = (const float*)d_in[5];
    const float* ti1  = (const float*)d_in[6];
    const float* ebr1 = (const float*)d_in[7];
    const float* ebi1 = (const float*)d_in[8];
    const float* Wr2  = (const float*)d_in[9];
    const float* Wi2  = (const float*)d_in[10];
    const float* br2  = (const float*)d_in[11];
    const float* bi2  = (const float*)d_in[12];
    const float* tr2  = (const float*)d_in[13];
    const float* ti2  = (const float*)d_in[14];
    const float* ebr2 = (const float*)d_in[15];
    const float* ebi2 = (const float*)d_in[16];
    const int*   ei   = (const int*)d_in[17];
    const int*   batch= (const int*)d_in[18];
    float* out = (float*)d_out;

    // workspace layout (offsets 256B aligned; total ~44.3 MB)
    char* ws = (char*)d_ws;
    unsigned* bitA  = (unsigned*)(ws + 0);          // 8 MB
    unsigned* bitAT = (unsigned*)(ws + 8388608);    // 8 MB
    float* G1r = (float*)(ws + 0);                  // reuse bitA after ba built
    float* G1i = (float*)(ws + 8388608);            // reuse bitAT
    float* ba  = (float*)(ws + 16777216);           // 8192*128 floats
    int*   sA  = (int*)  (ws + 20971520);           // 8192 ints
    float* u0c1= (float*)(ws + 21004288);           // 8192*4
    float* u0c2= (float*)(ws + 21135360);           // 8192*4
    float* m1r = (float*)(ws + 21266432);           // 256
    float* m1i = (float*)(ws + 21267456);           // 256
    float* m2r = (float*)(ws + 21268480);           // 128
    float* m2i = (float*)(ws + 21268992);           // 128
    float* H1r = (float*)(ws + 21269504);           // 8 MB
    float* H1i = (float*)(ws + 29658112);           // 8 MB
    float* G2r = (float*)(ws + 38046720);           // 4 MB
    float* G2i = (float*)(ws + 42241024);           // 4 MB
    float* xa  = (float*)(ws + 0);                  // reuse G1 region for |H2|

    // 1) zero adjacency bitsets (16 MB = 4M u32, contiguous)
    {
        int words = 2 * N * WPR;
        zero_u32_kernel<<<(words + 255) / 256, 256, 0, stream>>>(bitA, words);
    }
    // 2) edges -> bitsets
    build_bits_kernel<<<(E + 255) / 256, 256, 0, stream>>>(ei, bitA, bitAT, E);
    // 3) per-node 11x11 blocks + s
    build_ba_kernel<<<(N + 255) / 256, 256, 0, stream>>>(bitA, bitAT, ba, sA, N);
    // 4) matrix exponential per node, per layer (only needs ba, s, ct)
    expm_kernel<<<N / 8, 256, 0, stream>>>(ba, sA, tr1, ti1, u0c1);
    expm_kernel<<<N / 8, 256, 0, stream>>>(ba, sA, tr2, ti2, u0c2);
    // 5) layer 1 complex GEMM (real input): G1 = crelu(x@W1^T + b1)  [8192,256]
    cgemm_wmma_kernel<false><<<(N / 16) * (256 / 16) / 8, 256, 0, stream>>>(
        x, nullptr, Wr1, Wi1, br1, bi1, G1r, G1i, N, 128, 256);
    // 6) column means of G1
    colmean_kernel<<<2 * 256, 256, 0, stream>>>(G1r, G1i, m1r, m1i, N, 256);
    // 7) ev layer 1 -> H1
    ev_kernel<<<(N * 256) / 256, 256, 0, stream>>>(G1r, G1i, u0c1, m1r, m1i,
                                                   ebr1, ebi1, H1r, H1i, 256);
    // 8) layer 2 complex GEMM: G2 = crelu(H1@W2^T + b2)  [8192,128]
    cgemm_wmma_kernel<true><<<(N / 16) * (128 / 16) / 8, 256, 0, stream>>>(
        H1r, H1i, Wr2, Wi2, br2, bi2, G2r, G2i, N, 256, 128);
    // 9) column means of G2
    colmean_kernel<<<2 * 128, 256, 0, stream>>>(G2r, G2i, m2r, m2i, N, 128);
    // 10) ev layer 2 + abs -> xa (reuses G1 region, long dead)
    ev_abs_kernel<<<(N * 128) / 256, 256, 0, stream>>>(G2r, G2i, u0c2, m2r, m2i,
                                                       ebr2, ebi2, xa, 128);
    // 11) deterministic segment-mean pooling + log_softmax
    pool_softmax_kernel<<<NGRAPH, 128, 0, stream>>>(xa, batch, out, N, 128);
}